// CIKGRec_66958540145065
// MI455X (gfx1250) — compile-verified
//
#include <hip/hip_runtime.h>
#include <math.h>

#define N_USERS   100000
#define N_ENTITIES 400000
#define N_NODES   (N_USERS + N_ENTITIES)   // 500000
#define DIM       64
#define N_EDGES   2000000
#define BATCH     4096
#define N_LAYERS  3

typedef __attribute__((ext_vector_type(2))) float v2f;
typedef __attribute__((ext_vector_type(8))) float v8f;

// ---------------- reductions ----------------
__device__ __forceinline__ float blockReduceSum(float v) {
    __shared__ float smem[8];                      // <=256 threads / wave32
    #pragma unroll
    for (int o = 16; o > 0; o >>= 1) v += __shfl_down(v, o, 32);
    const int lane = threadIdx.x & 31, w = threadIdx.x >> 5;
    if (lane == 0) smem[w] = v;
    __syncthreads();
    const int nw = (blockDim.x + 31) >> 5;
    v = (threadIdx.x < nw) ? smem[threadIdx.x] : 0.0f;
    if (w == 0) {
        #pragma unroll
        for (int o = 4; o > 0; o >>= 1) v += __shfl_down(v, o, 32);
    }
    return v;  // valid in thread 0
}

__device__ __forceinline__ float vext8(v8f v, int i) {
    float r = v[0];
    #pragma unroll
    for (int j = 1; j < 8; ++j) r = (i == j) ? v[j] : r;
    return r;
}

// ---------------- degree / norm ----------------
__global__ void deg_kernel(const int* __restrict__ dst, float* __restrict__ deg, int nE) {
    int e = blockIdx.x * blockDim.x + threadIdx.x;
    if (e < nE) atomicAdd(&deg[dst[e]], 1.0f);
}

__global__ void dinv_kernel(const float* __restrict__ deg, float* __restrict__ dinv, int n) {
    int i = blockIdx.x * blockDim.x + threadIdx.x;
    if (i < n) {
        float d = deg[i];
        dinv[i] = (d > 0.0f) ? rsqrtf(fmaxf(d, 1e-12f)) : 0.0f;
    }
}

// ---------------- SpMM: 16 lanes per edge, float4 (b128) per lane ----------------
// 2 edges per wave; each half-wave streams one 256 B row fully coalesced.
__global__ void spmm_kernel(const float* __restrict__ xin, float* __restrict__ xout,
                            const int* __restrict__ src, const int* __restrict__ dst,
                            const float* __restrict__ dinv, int nE) {
    int gid = blockIdx.x * blockDim.x + threadIdx.x;
    int e   = gid >> 4;            // edge index
    int q   = gid & 15;            // quad-of-dims within the 64-dim row
    if (e >= nE) return;
    int s = src[e], d = dst[e];
    float nrm = dinv[s] * dinv[d];
    const float4 v = *(const float4*)(xin + (size_t)s * DIM + 4 * q);
    float* o = xout + (size_t)d * DIM + 4 * q;
    atomicAdd(o + 0, v.x * nrm);
    atomicAdd(o + 1, v.y * nrm);
    atomicAdd(o + 2, v.z * nrm);
    atomicAdd(o + 3, v.w * nrm);
}

// ---------------- batch gathers (layer-0 init + per-layer accumulate) ----------------
// 16 threads per batch row, float4 per thread.
__global__ void gather_init_kernel(const float* __restrict__ emb,
                                   const int* __restrict__ uidx, const int* __restrict__ pidx,
                                   const int* __restrict__ nidx,
                                   float* __restrict__ uacc, float* __restrict__ pacc,
                                   float* __restrict__ nacc, float* __restrict__ reg_sum) {
    int gid = blockIdx.x * blockDim.x + threadIdx.x;   // 3*BATCH*16 threads
    int r   = gid >> 4;
    int q   = gid & 15;
    int which = r / BATCH;
    int i     = r - which * BATCH;
    const int* idxA = (which == 0) ? uidx : ((which == 1) ? pidx : nidx);
    float*     acc  = (which == 0) ? uacc : ((which == 1) ? pacc : nacc);
    int node = idxA[i];
    float4 v = *(const float4*)(emb + (size_t)node * DIM + 4 * q);
    *(float4*)(acc + (size_t)i * DIM + 4 * q) = v;
    float sq = blockReduceSum(v.x * v.x + v.y * v.y + v.z * v.z + v.w * v.w);
    if (threadIdx.x == 0) atomicAdd(reg_sum, sq);
}

__global__ void gather_acc_kernel(const float* __restrict__ x,
                                  const int* __restrict__ uidx, const int* __restrict__ pidx,
                                  const int* __restrict__ nidx,
                                  float* __restrict__ uacc, float* __restrict__ pacc,
                                  float* __restrict__ nacc) {
    int gid = blockIdx.x * blockDim.x + threadIdx.x;
    int r   = gid >> 4;
    int q   = gid & 15;
    int which = r / BATCH;
    int i     = r - which * BATCH;
    const int* idxA = (which == 0) ? uidx : ((which == 1) ? pidx : nidx);
    float*     acc  = (which == 0) ? uacc : ((which == 1) ? pacc : nacc);
    int node = idxA[i];
    float4 v = *(const float4*)(x + (size_t)node * DIM + 4 * q);
    float4* a = (float4*)(acc + (size_t)i * DIM + 4 * q);
    float4 cur = *a;
    cur.x += v.x; cur.y += v.y; cur.z += v.z; cur.w += v.w;
    *a = cur;
}

// ---------------- WMMA scoring: diag of 16x16 tiles, K=64 via 16x16x4_f32 ----------------
// One wave per tile of 16 batch rows. A-layout (16x4 f32): lane m<16 holds K=0,1;
// lane m+16 holds K=2,3. B-layout (4x16): VGPR0 = rows {0|2}, VGPR1 = rows {1|3}.
// Both reduce to the same float2 load: row = tile*16+(lane&15), col = 4k + 2*(lane>>4).
__global__ void score_wmma_kernel(const float* __restrict__ uacc, const float* __restrict__ pacc,
                                  const float* __restrict__ nacc,
                                  float* __restrict__ pos_s, float* __restrict__ neg_s) {
    const int tile = blockIdx.x;          // 256 tiles of 16 rows
    const int lane = threadIdx.x;         // blockDim == 32, EXEC all ones
    const int m    = lane & 15;
    const int h    = lane >> 4;
    const size_t row = (size_t)(tile * 16 + m) * DIM;
    v8f cp = {}, cn = {};
    #pragma unroll
    for (int k = 0; k < 16; ++k) {
        const int col = 4 * k + 2 * h;
        v2f a  = *(const v2f*)(uacc + row + col);
        v2f bp = *(const v2f*)(pacc + row + col);
        v2f bn = *(const v2f*)(nacc + row + col);
        cp = __builtin_amdgcn_wmma_f32_16x16x4_f32(false, a, false, bp, (short)0, cp, false, false);
        cn = __builtin_amdgcn_wmma_f32_16x16x4_f32(false, a, false, bn, (short)0, cn, false, false);
    }
    const float scale = 1.0f / 16.0f;     // (acc/4)·(acc/4)
    if (lane < 8) {                        // diag rows 0..7: VGPR i, lane i
        pos_s[tile * 16 + lane] = vext8(cp, lane) * scale;
        neg_s[tile * 16 + lane] = vext8(cn, lane) * scale;
    } else if (lane >= 24) {               // diag rows 8..15: VGPR i-8, lane i+16
        pos_s[tile * 16 + lane - 16] = vext8(cp, lane - 24) * scale;
        neg_s[tile * 16 + lane - 16] = vext8(cn, lane - 24) * scale;
    }
}

// ---------------- loss reduce + finalize ----------------
__global__ void loss_kernel(const float* __restrict__ pos_s, const float* __restrict__ neg_s,
                            float* __restrict__ cf_sum) {
    int i = blockIdx.x * blockDim.x + threadIdx.x;
    float v = 0.0f;
    if (i < BATCH) {
        float x = neg_s[i] - pos_s[i];
        v = fmaxf(x, 0.0f) + log1pf(expf(-fabsf(x)));   // stable softplus
    }
    v = blockReduceSum(v);
    if (threadIdx.x == 0) atomicAdd(cf_sum, v);
}

__global__ void finalize_kernel(const float* __restrict__ cf_sum,
                                const float* __restrict__ reg_sum, float* __restrict__ out) {
    out[0] = cf_sum[0] / (float)BATCH + 1e-4f * (0.5f * reg_sum[0] / (float)BATCH);
}

// ---------------- host launcher ----------------
extern "C" void kernel_launch(void* const* d_in, const int* in_sizes, int n_in,
                              void* d_out, int out_size, void* d_ws, size_t ws_size,
                              hipStream_t stream) {
    const float* emb  = (const float*)d_in[0];
    const int*   eidx = (const int*)d_in[1];
    const int*   src  = eidx;
    const int*   dst  = eidx + N_EDGES;
    const int*   uidx = (const int*)d_in[2];
    const int*   pidx = (const int*)d_in[3];
    const int*   nidx = (const int*)d_in[4];
    float*       out  = (float*)d_out;

    char* ws = (char*)d_ws;
    size_t off = 0;
    auto take = [&](size_t bytes) -> char* {
        char* p = ws + off;
        off += (bytes + 255) & ~(size_t)255;
        return p;
    };
    const size_t XBYTES = (size_t)N_NODES * DIM * sizeof(float);   // 128 MB
    float* deg   = (float*)take((size_t)N_NODES * sizeof(float));
    float* dinv  = (float*)take((size_t)N_NODES * sizeof(float));
    float* xA    = (float*)take(XBYTES);
    float* xB    = (float*)take(XBYTES);
    float* uacc  = (float*)take((size_t)BATCH * DIM * sizeof(float));
    float* pacc  = (float*)take((size_t)BATCH * DIM * sizeof(float));
    float* nacc  = (float*)take((size_t)BATCH * DIM * sizeof(float));
    float* pos_s = (float*)take((size_t)BATCH * sizeof(float));
    float* neg_s = (float*)take((size_t)BATCH * sizeof(float));
    float* cf_sum  = (float*)take(sizeof(float));
    float* reg_sum = (float*)take(sizeof(float));

    hipMemsetAsync(deg, 0, (size_t)N_NODES * sizeof(float), stream);
    hipMemsetAsync(cf_sum, 0, sizeof(float), stream);
    hipMemsetAsync(reg_sum, 0, sizeof(float), stream);

    // degree + normalization
    deg_kernel<<<(N_EDGES + 255) / 256, 256, 0, stream>>>(dst, deg, N_EDGES);
    dinv_kernel<<<(N_NODES + 255) / 256, 256, 0, stream>>>(deg, dinv, N_NODES);

    // layer-0 term of the mean + L2 reg on ego embeddings
    const int gthreads = 3 * BATCH * 16;
    gather_init_kernel<<<gthreads / 256, 256, 0, stream>>>(emb, uidx, pidx, nidx,
                                                           uacc, pacc, nacc, reg_sum);

    // 3 propagation layers; accumulate only the gathered batch rows
    const float* xin = emb;
    for (int l = 0; l < N_LAYERS; ++l) {
        float* xout = (l & 1) ? xB : xA;   // l0->A, l1->B, l2->A
        hipMemsetAsync(xout, 0, XBYTES, stream);
        spmm_kernel<<<((size_t)N_EDGES * 16) / 256, 256, 0, stream>>>(xin, xout, src, dst,
                                                                      dinv, N_EDGES);
        gather_acc_kernel<<<gthreads / 256, 256, 0, stream>>>(xout, uidx, pidx, nidx,
                                                              uacc, pacc, nacc);
        xin = xout;
    }

    // scores via fp32 WMMA tiles, then BPR loss
    score_wmma_kernel<<<BATCH / 16, 32, 0, stream>>>(uacc, pacc, nacc, pos_s, neg_s);
    loss_kernel<<<(BATCH + 255) / 256, 256, 0, stream>>>(pos_s, neg_s, cf_sum);
    finalize_kernel<<<1, 1, 0, stream>>>(cf_sum, reg_sum, out);
}